// AttentionSE3_6433861009743
// MI455X (gfx1250) — compile-verified
//
#include <hip/hip_runtime.h>

#define N_NODES   50000
#define N_EDGES   800000
#define NUM_HEADS 8
#define HEAD_DIM  32
#define VAL_CH    64
#define VAL_DIM   4

typedef __attribute__((ext_vector_type(2))) float v2f;
typedef __attribute__((ext_vector_type(4))) float v4f;
typedef __attribute__((ext_vector_type(8))) float v8f;

// ---------- helpers ----------

// order-preserving float <-> uint encoding (for atomic max on f32)
__device__ __forceinline__ unsigned ford(float f) {
    int i = __float_as_int(f);
    return (i >= 0) ? ((unsigned)i ^ 0x80000000u) : ~(unsigned)i;
}
__device__ __forceinline__ float funord(unsigned u) {
    int i = (u & 0x80000000u) ? (int)(u ^ 0x80000000u) : (int)~u;
    return __int_as_float(i);
}

// native f32 global atomic add (no CAS fallback)
__device__ __forceinline__ void atom_add_f32(float* p, float v) {
    asm volatile("global_atomic_add_f32 %0, %1, off" : : "v"(p), "v"(v) : "memory");
}

// ---------- kernel 0: zero output + init segment buffers ----------
__global__ __launch_bounds__(256) void k_init(float* __restrict__ out,
                                              float* __restrict__ segsum,
                                              unsigned* __restrict__ segmax) {
    int i = blockIdx.x * 256 + threadIdx.x;              // grid covers N*256
    out[i] = 0.0f;                                       // N_NODES*64*4 elements
    if (i < N_NODES * NUM_HEADS) {
        segsum[i] = 0.0f;
        segmax[i] = 0u;                                  // below encoding of any finite float
    }
}

// ---------- kernel 1: scores via V_WMMA_F32_16X16X4_F32 + atomic segment max ----------
// Block = 256 threads = 8 waves; wave h handles head h of a 16-edge tile.
// A = K-tile (16 edges x 4 K), B = gathered Q^T (4 K x 16 edges); after 8 chained
// WMMAs over K=32, diag(D) holds the 16 per-edge dot products.
//
// B gather is branchless: for lane column n, head h, iteration t = kc/4 the
// concat-query element q[n,h,kc+kb2+{0,1}] lives at channel c = n*64 + h*8 + t with
// fixed sub-rep j per lane half (j={0,1} lanes 0-15 from q0/q1, j={2,3} lanes 16-31
// both from q1). So two base pointers selected once, advanced by constant strides.
__global__ __launch_bounds__(256) void k_scores(const float* __restrict__ key_edge,
                                                const float* __restrict__ q0,
                                                const float* __restrict__ q1,
                                                const int* __restrict__ dst,
                                                float* __restrict__ scores,
                                                unsigned* __restrict__ segmax) {
    const int lane = threadIdx.x & 31;
    const int h    = threadIdx.x >> 5;                   // head (8 waves/block)
    const int e0   = blockIdx.x * 16;                    // edge tile base
    const int r    = lane & 15;                          // row / column (edge within tile)
    const int kb2  = (lane >> 4) * 2;                    // lanes 0-15 -> K+{0,1}, 16-31 -> K+{2,3}

    const int    e    = e0 + r;
    const int    n    = dst[e];
    const float* krow = key_edge + ((long)e * NUM_HEADS + h) * HEAD_DIM;

    // branchless per-lane B-operand base pointers + strides
    const int    t0 = n * 64 + h * 8;                    // channel base for this column
    const bool   hi = (kb2 != 0);
    const float* p1 = q1 + (long)t0 * 3;
    const float* px = hi ? (p1 + 1) : (q0 + t0);         // j = kb2
    const float* py = hi ? (p1 + 2) : p1;                // j = kb2 + 1
    const int    sx = hi ? 3 : 1;

    v8f acc = {};
#pragma unroll
    for (int kc = 0; kc < HEAD_DIM; kc += 4) {
        // A: key_edge streamed once -> non-temporal
        v2f a = __builtin_nontemporal_load((const v2f*)(krow + kc + kb2));
        // B: gathered query (hot, keep cached in L2)
        v2f b;
        b.x = *px;
        b.y = *py;
        px += sx;
        py += 3;
        acc = __builtin_amdgcn_wmma_f32_16x16x4_f32(
            /*neg_a=*/false, a, /*neg_b=*/false, b,
            /*c_mod=*/(short)0, acc, /*reuse_a=*/false, /*reuse_b=*/false);
    }

    // diagonal extraction: row i<8 at (vgpr i, lane i); row i>=8 at (vgpr i-8, lane i+16)
    int row = -1, idx = 0;
    if (lane < 8)        { row = lane;      idx = lane;      }
    else if (lane >= 24) { row = lane - 16; idx = lane - 24; }

    float s = acc[0];
#pragma unroll
    for (int i = 1; i < 8; ++i)
        if (idx == i) s = acc[i];

    if (row >= 0) {
        float sc = s * 0.0625f;                          // 1/sqrt(256)
        int   er = e0 + row;
        scores[er * NUM_HEADS + h] = sc;
        atomicMax(&segmax[dst[er] * NUM_HEADS + h], ford(sc));
    }
}

// ---------- kernel 2: ex = exp(score - max), atomic segment sum ----------
__global__ __launch_bounds__(256) void k_softmax(float* __restrict__ ex,
                                                 const int* __restrict__ dst,
                                                 const unsigned* __restrict__ segmax,
                                                 float* __restrict__ segsum) {
    int t = blockIdx.x * 256 + threadIdx.x;              // E*H threads
    int e = t >> 3;
    int h = t & 7;
    int n = dst[e];
    float m = funord(segmax[n * NUM_HEADS + h]);
    float v = __expf(ex[t] - m);
    ex[t] = v;
    atom_add_f32(&segsum[n * NUM_HEADS + h], v);
}

// ---------- kernel 3: weighted scatter of values ----------
// One thread per (edge, channel): float4 NT load of value[e,c,:], 4 f32 atomic adds
// into the L2-resident 51 MB output region.
__global__ __launch_bounds__(256) void k_scatter(const float* __restrict__ value,
                                                 const float* __restrict__ ex,
                                                 const float* __restrict__ segsum,
                                                 const int* __restrict__ dst,
                                                 float* __restrict__ out) {
    long t = (long)blockIdx.x * 256 + threadIdx.x;       // E*64 threads
    int  e = (int)(t >> 6);
    int  c = (int)t & 63;
    int  h = c >> 3;
    int  n = dst[e];
    float w = ex[e * NUM_HEADS + h] / segsum[n * NUM_HEADS + h];
    v4f  v  = __builtin_nontemporal_load((const v4f*)(value + (t << 2)));   // value streamed once
    float* o = out + ((long)n * 64 + c) * 4;
    atom_add_f32(o + 0, w * v.x);
    atom_add_f32(o + 1, w * v.y);
    atom_add_f32(o + 2, w * v.z);
    atom_add_f32(o + 3, w * v.w);
}

// ---------- launcher ----------
extern "C" void kernel_launch(void* const* d_in, const int* in_sizes, int n_in,
                              void* d_out, int out_size, void* d_ws, size_t ws_size,
                              hipStream_t stream) {
    const float* key_edge = (const float*)d_in[0];   // [E, 8, 32]
    const float* q0       = (const float*)d_in[1];   // [N, 64, 1]
    const float* q1       = (const float*)d_in[2];   // [N, 64, 3]
    const float* value    = (const float*)d_in[3];   // [E, 64, 4]
    const int*   dst      = (const int*)d_in[4];     // [E]
    float*       out      = (float*)d_out;           // [N, 64, 4]

    char* ws = (char*)d_ws;
    float*    ex     = (float*)ws;                                       // E*H floats (25.6 MB)
    unsigned* segmax = (unsigned*)(ws + sizeof(float) * (size_t)N_EDGES * NUM_HEADS);
    float*    segsum = (float*)((char*)segmax + sizeof(unsigned) * (size_t)N_NODES * NUM_HEADS);

    k_init   <<<(N_NODES * VAL_CH * VAL_DIM) / 256, 256, 0, stream>>>(out, segsum, segmax);
    k_scores <<<N_EDGES / 16,                      256, 0, stream>>>(key_edge, q0, q1, dst, ex, segmax);
    k_softmax<<<(N_EDGES * NUM_HEADS) / 256,       256, 0, stream>>>(ex, dst, segmax, segsum);
    k_scatter<<<((size_t)N_EDGES * VAL_CH) / 256,  256, 0, stream>>>(value, ex, segsum, dst, out);
}